// TQUnitaryBuilder_62216896250328
// MI455X (gfx1250) — compile-verified
//
#include <hip/hip_runtime.h>
#include <hip/hip_bf16.h>

// ---------------------------------------------------------------------------
// TQUnitaryBuilder, MI455X (gfx1250, wave32).
//  out[b] = U_stack[b] @ expm(A_skew[b])  reduced algebraically to
//  out[:,i] = col_i + c1(i)*u0 + c2(i)*w   (rank-2 epilogue, exact:
//  S = e0 a^T - a e0^T, S^3 = -theta^2 S  =>  expm(S) = I + a*S + b*S^2)
//  Statevector-per-wave simulation: 16 VGPR re + 16 VGPR im per lane.
// ---------------------------------------------------------------------------

typedef __attribute__((ext_vector_type(2))) float v2f;
typedef __attribute__((ext_vector_type(8))) float v8f;

#define BATCH   64
#define EMBED   512
#define PGATE   72      // 9 wires * 4 layers * 2
#define DDIM    512

struct SV { float re[16]; float im[16]; };

__device__ __forceinline__ float shflx(float v, int m) { return __shfl_xor(v, m, 32); }

// silu via raw v_exp + v_rcp (avoid IEEE div expansion in the WMMA loop)
__device__ __forceinline__ float fast_silu(float x) {
  return x * __builtin_amdgcn_rcpf(1.f + __expf(-x));
}

// Fused Rz@Ry gate: g00=(er-i ei)cy, g01=-(er-i ei)sy, g10=(er+i ei)sy, g11=(er+i ei)cy
// Wire on a register bit RB (amplitude bits 8..5 -> reg bits 3..0).
template<int RB>
__device__ __forceinline__ void gate_reg(SV& s, float cy, float sy, float er, float ei) {
  constexpr int m = 1 << RB;
#pragma unroll
  for (int r = 0; r < 16; ++r) {
    if (r & m) continue;
    const int r1 = r | m;
    float x0r = s.re[r],  x0i = s.im[r];
    float x1r = s.re[r1], x1i = s.im[r1];
    float ur = cy * x0r - sy * x1r, ui = cy * x0i - sy * x1i;
    float vr = sy * x0r + cy * x1r, vi = sy * x0i + cy * x1i;
    s.re[r]  = er * ur + ei * ui;   // (er - i ei) * u
    s.im[r]  = er * ui - ei * ur;
    s.re[r1] = er * vr - ei * vi;   // (er + i ei) * v
    s.im[r1] = er * vi + ei * vr;
  }
}

// Wire on a lane bit LB (amplitude bits 4..0 -> lane bits 4..0). Cross-lane pairs.
template<int LB>
__device__ __forceinline__ void gate_lane(SV& s, int lane, float cy, float sy, float er, float ei) {
  const int m = 1 << LB;
  const bool hi = (lane & m) != 0;
  // low lane:  y = (er - i ei)(cy*x - sy*p) ; high lane: y = (er + i ei)(cy*x + sy*p)
  const float st = hi ?  sy : -sy;
  const float se = hi ?  ei : -ei;
#pragma unroll
  for (int r = 0; r < 16; ++r) {
    float xr = s.re[r], xi = s.im[r];
    float pr = shflx(xr, m), pi = shflx(xi, m);
    float wr = cy * xr + st * pr;
    float wi = cy * xi + st * pi;
    s.re[r] = er * wr - se * wi;
    s.im[r] = er * wi + se * wr;
  }
}

// CNOT: control/target both on register bits — pure register relabel.
template<int RC, int RT>
__device__ __forceinline__ void cnot_rr(SV& s) {
#pragma unroll
  for (int r = 0; r < 16; ++r) {
    if (((r >> RC) & 1) && !((r >> RT) & 1)) {
      const int r1 = r | (1 << RT);
      float t = s.re[r]; s.re[r] = s.re[r1]; s.re[r1] = t;
      t = s.im[r]; s.im[r] = s.im[r1]; s.im[r1] = t;
    }
  }
}

// CNOT: control on register bit, target on lane bit — shuffle the control=1 regs.
template<int RC, int LT>
__device__ __forceinline__ void cnot_rl(SV& s) {
#pragma unroll
  for (int r = 0; r < 16; ++r) {
    if ((r >> RC) & 1) {
      s.re[r] = shflx(s.re[r], 1 << LT);
      s.im[r] = shflx(s.im[r], 1 << LT);
    }
  }
}

// CNOT: control and target on lane bits — per-lane select after shuffle.
template<int LC, int LT>
__device__ __forceinline__ void cnot_ll(SV& s, int lane) {
  const bool c = (lane & (1 << LC)) != 0;
#pragma unroll
  for (int r = 0; r < 16; ++r) {
    float pr = shflx(s.re[r], 1 << LT);
    float pi = shflx(s.im[r], 1 << LT);
    s.re[r] = c ? pr : s.re[r];
    s.im[r] = c ? pi : s.im[r];
  }
}

// CNOT: control on lane bit, target on register bit — conditional reg-pair swap.
template<int LC, int RT>
__device__ __forceinline__ void cnot_lr(SV& s, int lane) {
  const bool c = (lane & (1 << LC)) != 0;
#pragma unroll
  for (int r = 0; r < 16; ++r) {
    if (!((r >> RT) & 1)) {
      const int r1 = r | (1 << RT);
      float a = s.re[r], b = s.re[r1];
      s.re[r] = c ? b : a; s.re[r1] = c ? a : b;
      a = s.im[r]; b = s.im[r1];
      s.im[r] = c ? b : a; s.im[r1] = c ? a : b;
    }
  }
}

// Full 4-layer circuit. Wire w acts on amplitude bit (8-w).
// gt: per-batch gate table, [layer][wire][cy,sy,er,ei] = 4*36 floats (uniform addr -> s_load).
__device__ __forceinline__ void simulate(SV& s, const float* __restrict__ gt, int lane) {
  for (int L = 0; L < 4; ++L) {
    const float* g = gt + L * 36;
    gate_reg<3>(s, g[0],  g[1],  g[2],  g[3]);    // wire 0 -> bit 8
    gate_reg<2>(s, g[4],  g[5],  g[6],  g[7]);    // wire 1 -> bit 7
    gate_reg<1>(s, g[8],  g[9],  g[10], g[11]);   // wire 2 -> bit 6
    gate_reg<0>(s, g[12], g[13], g[14], g[15]);   // wire 3 -> bit 5
    gate_lane<4>(s, lane, g[16], g[17], g[18], g[19]); // wire 4 -> bit 4
    gate_lane<3>(s, lane, g[20], g[21], g[22], g[23]); // wire 5
    gate_lane<2>(s, lane, g[24], g[25], g[26], g[27]); // wire 6
    gate_lane<1>(s, lane, g[28], g[29], g[30], g[31]); // wire 7
    gate_lane<0>(s, lane, g[32], g[33], g[34], g[35]); // wire 8 -> bit 0
    cnot_rr<3, 2>(s);          // CNOT(0,1): bits (8,7)
    cnot_rr<2, 1>(s);          // CNOT(1,2)
    cnot_rr<1, 0>(s);          // CNOT(2,3)
    cnot_rl<0, 4>(s);          // CNOT(3,4): reg0 -> lane4
    cnot_ll<4, 3>(s, lane);    // CNOT(4,5)
    cnot_ll<3, 2>(s, lane);    // CNOT(5,6)
    cnot_ll<2, 1>(s, lane);    // CNOT(6,7)
    cnot_ll<1, 0>(s, lane);    // CNOT(7,8)
    cnot_lr<0, 3>(s, lane);    // CNOT(8,0): lane0 -> reg3
  }
}

// ---------------------------------------------------------------------------
// K1: params = silu(t[:,None]*w1 + b1) @ w2 + b2  (64x512x72), f32 WMMA 16x16x4.
// 20 waves = 4 M-tiles x 5 N-tiles (N padded 72->80), K loop of 128 steps.
// ---------------------------------------------------------------------------
__global__ __launch_bounds__(640) void k_params(
    const float* __restrict__ t,  const float* __restrict__ w1,
    const float* __restrict__ b1, const float* __restrict__ w2,
    const float* __restrict__ b2, float* __restrict__ params) {
  const int tid  = threadIdx.x;
  const int wave = tid >> 5, lane = tid & 31;
  const int mt = wave / 5, nt = wave % 5;
  const int row   = mt * 16 + (lane & 15);
  const int n     = nt * 16 + (lane & 15);
  const int khalf = (lane >> 4) * 2;            // A/B K-split across lane halves
  const float tv = t[row];
  v8f acc = {};
  for (int k = 0; k < EMBED; k += 4) {
    const int k0 = k + khalf, k1 = k0 + 1;
    v2f a, b;
    a.x = fast_silu(tv * w1[k0] + b1[k0]);
    a.y = fast_silu(tv * w1[k1] + b1[k1]);
    b.x = (n < PGATE) ? w2[k0 * PGATE + n] : 0.f;
    b.y = (n < PGATE) ? w2[k1 * PGATE + n] : 0.f;
    acc = __builtin_amdgcn_wmma_f32_16x16x4_f32(false, a, false, b,
                                                (short)0, acc, false, false);
  }
  // D layout: VGPR v, lanes0-15 -> M=v, lanes16-31 -> M=v+8, N=lane&15
  const int mbase = mt * 16 + ((lane >= 16) ? 8 : 0);
  if (n < PGATE) {
#pragma unroll
    for (int v = 0; v < 8; ++v)
      params[(mbase + v) * PGATE + n] = acc[v] + b2[n];
  }
}

// ---------------------------------------------------------------------------
// K2: per-batch gate table (cy,sy,er,ei per gate) + alpha/beta/theta^2.
// ---------------------------------------------------------------------------
__global__ __launch_bounds__(64) void k_derive(
    const float* __restrict__ params, float* __restrict__ gtab, float* __restrict__ sc) {
  const int b = blockIdx.x, tid = threadIdx.x;
  const float* p = params + b * PGATE;
  if (tid < 36) {
    float thy = p[2 * tid], thz = p[2 * tid + 1];
    float* g = gtab + b * 144 + tid * 4;
    g[0] = cosf(0.5f * thy); g[1] = sinf(0.5f * thy);
    g[2] = cosf(0.5f * thz); g[3] = sinf(0.5f * thz);
  }
  float part = 0.f;
  for (int i = tid; i < PGATE; i += 64)
    if (i >= 1) { float a = 0.01f * p[i]; part += a * a; }
  __shared__ float red[64];
  red[tid] = part;
  __syncthreads();
  if (tid == 0) {
    float t2 = 0.f;
    for (int i = 0; i < 64; ++i) t2 += red[i];
    float th = sqrtf(t2), al, be;
    if (th > 1e-4f) { al = sinf(th) / th; be = (1.f - cosf(th)) / t2; }
    else            { al = 1.f - t2 / 6.f; be = 0.5f - t2 / 24.f; }
    sc[b * 4 + 0] = al; sc[b * 4 + 1] = be; sc[b * 4 + 2] = t2;
  }
}

// ---------------------------------------------------------------------------
// K3: per batch, simulate u0 = Re(U) e0 (wave 0) and w = Re(U) a (wave 1).
// ---------------------------------------------------------------------------
__global__ __launch_bounds__(64) void k_u0w(
    const float* __restrict__ params, const float* __restrict__ gtab,
    float* __restrict__ u0, float* __restrict__ wvec) {
  const int b = blockIdx.x;
  const int wv = threadIdx.x >> 5, lane = threadIdx.x & 31;
  SV s;
#pragma unroll
  for (int r = 0; r < 16; ++r) {
    const int j = r * 32 + lane;
    s.im[r] = 0.f;
    if (wv == 0) s.re[r] = (j == 0) ? 1.f : 0.f;
    else         s.re[r] = (j >= 1 && j < PGATE) ? 0.01f * params[b * PGATE + j] : 0.f;
  }
  simulate(s, gtab + b * 144, lane);
  float* dst = (wv == 0 ? u0 : wvec) + b * DDIM;
#pragma unroll
  for (int r = 0; r < 16; ++r) dst[r * 32 + lane] = s.re[r];
}

// ---------------------------------------------------------------------------
// K4: main. Block = 16 waves = 16 columns of one batch. Each wave simulates
// column i = cg*16+wv fully in registers, applies the exact rank-2 expm
// epilogue, stages 512x16 f32 in LDS (stride 17, bank-conflict-free), then
// the block writes 64B-contiguous rows to d_out.
// ---------------------------------------------------------------------------
__global__ __launch_bounds__(512) void k_main(
    const float* __restrict__ params, const float* __restrict__ gtab,
    const float* __restrict__ sc, const float* __restrict__ u0,
    const float* __restrict__ wvec, float* __restrict__ out) {
  const int cg = blockIdx.x, b = blockIdx.y;
  const int wv = threadIdx.x >> 5, lane = threadIdx.x & 31;
  const int i = cg * 16 + wv;                    // column (input basis index)
  __builtin_prefetch(gtab + b * 144, 0, 1);      // global_prefetch gate table
  SV s;
#pragma unroll
  for (int r = 0; r < 16; ++r) {
    s.im[r] = 0.f;
    s.re[r] = ((r * 32 + lane) == i) ? 1.f : 0.f;
  }
  simulate(s, gtab + b * 144, lane);

  const float al = sc[b * 4], be = sc[b * 4 + 1], t2 = sc[b * 4 + 2];
  const float ai = (i >= 1 && i < PGATE) ? 0.01f * params[b * PGATE + i] : 0.f;
  const float c1 = al * ai - ((i == 0) ? be * t2 : 0.f);
  const float c2 = -(((i == 0) ? al : 0.f) + be * ai);
  const float* u0p = u0 + b * DDIM;
  const float* wp  = wvec + b * DDIM;

  __shared__ float lds[DDIM * 17];
#pragma unroll
  for (int r = 0; r < 16; ++r) {
    const int j = r * 32 + lane;
    lds[j * 17 + wv] = s.re[r] + c1 * u0p[j] + c2 * wp[j];
  }
  __syncthreads();

  const int tid = threadIdx.x;
#pragma unroll
  for (int it = 0; it < 4; ++it) {
    const int j = it * 128 + (tid >> 2);
    const int c = (tid & 3) * 4;
    float4 v;
    v.x = lds[j * 17 + c + 0];
    v.y = lds[j * 17 + c + 1];
    v.z = lds[j * 17 + c + 2];
    v.w = lds[j * 17 + c + 3];
    float4* dst = (float4*)(out + ((size_t)(b * DDIM + j)) * DDIM + cg * 16 + c);
    *dst = v;
  }
}

// ---------------------------------------------------------------------------
extern "C" void kernel_launch(void* const* d_in, const int* in_sizes, int n_in,
                              void* d_out, int out_size, void* d_ws, size_t ws_size,
                              hipStream_t stream) {
  (void)in_sizes; (void)n_in; (void)out_size; (void)ws_size;
  const float* t  = (const float*)d_in[0];
  const float* w1 = (const float*)d_in[1];
  const float* b1 = (const float*)d_in[2];
  const float* w2 = (const float*)d_in[3];
  const float* b2 = (const float*)d_in[4];
  float* out = (float*)d_out;
  float* ws  = (float*)d_ws;
  float* params = ws;             // 64*72   = 4608 f
  float* gtab   = ws + 4608;      // 64*144  = 9216 f
  float* sc     = ws + 13824;     // 64*4    = 256 f
  float* u0     = ws + 14080;     // 64*512  = 32768 f
  float* wvec   = ws + 46848;     // 64*512  = 32768 f  (total ~311 KB)

  k_params<<<dim3(1), dim3(640), 0, stream>>>(t, w1, b1, w2, b2, params);
  k_derive<<<dim3(BATCH), dim3(64), 0, stream>>>(params, gtab, sc);
  k_u0w<<<dim3(BATCH), dim3(64), 0, stream>>>(params, gtab, u0, wvec);
  k_main<<<dim3(DDIM / 16, BATCH), dim3(512), 0, stream>>>(params, gtab, sc, u0, wvec, out);
}